// GraphGather_12721693131106
// MI455X (gfx1250) — compile-verified
//
#include <hip/hip_runtime.h>

// Segment-sum (sorted keys) via V_WMMA_F32_16X16X4_F32 on gfx1250 (MI455X).
//
// inputs : d_in[0] = atom_features (524288 x 128) fp32
//          d_in[1] = membership    (524288)       int32, sorted ascending
//          d_in[2] = batch_size scalar (int)
// output : d_out = (batch_size x 128) fp32
//
// Memory-bound: 268 MB stream => ~12 us floor @ 23.3 TB/s. The per-group
// scatter-reduce runs on the matrix pipe:
//   - ballot over "membership changed" -> run-rank per atom (SGPR mask math,
//     no LDS shuffles)
//   - A = one-hot(rank x atom) 16x4 fp32 built in registers
//   - B = features (4 atoms x 16 feats), NT loads
//   - C += A*B over 4 K-steps => C row r = partial sum of local run r
//   - flush live rows with global_atomic_add_f32 (avg ~1.5 rows / 16 atoms)

typedef __attribute__((ext_vector_type(2))) float v2f;
typedef __attribute__((ext_vector_type(8))) float v8f;
typedef __attribute__((ext_vector_type(4))) float f4;

#define NFEAT 128
#define GROUP 16      // atoms per WMMA group
#define GPB   8       // groups per block
#define ATOMS_PER_BLOCK (GROUP * GPB)   // 128

__device__ __forceinline__ float row_val(const v8f& c, int r) {
    switch (r) {
        case 0: return c[0]; case 1: return c[1];
        case 2: return c[2]; case 3: return c[3];
        case 4: return c[4]; case 5: return c[5];
        case 6: return c[6]; default: return c[7];
    }
}

__global__ void __launch_bounds__(256) zero_out_kernel(float* __restrict__ out, int n4) {
    int i = blockIdx.x * blockDim.x + threadIdx.x;
    if (i < n4) ((f4*)out)[i] = f4{0.f, 0.f, 0.f, 0.f};
}

__global__ void __launch_bounds__(256) segsum_wmma_kernel(
    const float* __restrict__ feats,
    const int*   __restrict__ mem,
    float*       __restrict__ out,
    int n_atoms)
{
    const int lane = threadIdx.x & 31;
    const int wave = threadIdx.x >> 5;      // 8 waves: one 16-feature slice each
    const int n    = lane & 15;             // N (feature) / M (rank) / atom index
    const int half = lane >> 4;             // selects K pair & D row half
    const int featBase = wave * 16;

    for (int grp = 0; grp < GPB; ++grp) {
        const int atomBase = blockIdx.x * ATOMS_PER_BLOCK + grp * GROUP;
        if (atomBase >= n_atoms) return;

        // --- run-boundary ballot (halves mirror; mask lives in an SGPR)
        const int mm    = mem[atomBase + n];
        const int mprev = __shfl_up(mm, 1, 16);
        const unsigned changed = (n != 0 && mm != mprev) ? 1u : 0u;
        const unsigned blow = __builtin_amdgcn_ballot_w32((int)changed) & 0xFFFFu;

        // --- D = sum over 4 K-steps of onehot(A) x feats(B), C starts at 0.
        //     rank(atom k) = popc(blow & ((1<<k)-1))  -- pure SGPR/VALU math,
        //     no cross-lane traffic needed to build A.
        v8f c = {0.f,0.f,0.f,0.f,0.f,0.f,0.f,0.f};
        #pragma unroll
        for (int sub = 0; sub < 4; ++sub) {
            const int kx = sub * 4 + 2 * half;   // atom feeding A/B component .x
            const int ky = kx + 1;               // atom feeding A/B component .y
            const int lx = __popc(blow & ((1u << kx) - 1u));
            const int ly = __popc(blow & ((1u << ky) - 1u));
            v2f a, b;
            a.x = (lx == n) ? 1.0f : 0.0f;
            a.y = (ly == n) ? 1.0f : 0.0f;
            b.x = __builtin_nontemporal_load(
                    feats + (size_t)(atomBase + kx) * NFEAT + featBase + n);
            b.y = __builtin_nontemporal_load(
                    feats + (size_t)(atomBase + ky) * NFEAT + featBase + n);
            c = __builtin_amdgcn_wmma_f32_16x16x4_f32(
                    /*neg_a=*/false, a, /*neg_b=*/false, b,
                    /*c_mod=*/(short)0, c, /*reuse_a=*/false, /*reuse_b=*/false);
        }

        // --- flush live rows (r, idx, bits are wave-uniform; readlane broadcast)
        const int nseg = __popc(blow) + 1;
        unsigned bits = blow;
        int idx = 0;                              // atom index starting run r
        for (int r = 0; r < nseg; ++r) {
            const int   seg = __builtin_amdgcn_readlane(mm, idx);
            const float v   = row_val(c, r & 7);
            if (half == (r >> 3))                 // row r lives in this lane half
                atomicAdd(out + (size_t)seg * NFEAT + featBase + n, v);
            idx  = __ffs((int)bits) - 1;          // start of next run
            bits &= bits - 1;
        }
    }
}

extern "C" void kernel_launch(void* const* d_in, const int* in_sizes, int n_in,
                              void* d_out, int out_size, void* d_ws, size_t ws_size,
                              hipStream_t stream) {
    const float* feats = (const float*)d_in[0];
    const int*   mem   = (const int*)d_in[1];
    float*       out   = (float*)d_out;

    const int n_atoms = in_sizes[0] / NFEAT;      // 524288

    // Zero the output (we accumulate with atomics; harness poisons d_out).
    const int n4 = out_size / 4;
    zero_out_kernel<<<(n4 + 255) / 256, 256, 0, stream>>>(out, n4);

    const int n_blocks = (n_atoms + ATOMS_PER_BLOCK - 1) / ATOMS_PER_BLOCK; // 4096
    segsum_wmma_kernel<<<n_blocks, 256, 0, stream>>>(feats, mem, out, n_atoms);
}